// SamplesLoss_61272003445419
// MI455X (gfx1250) — compile-verified
//
#include <hip/hip_runtime.h>
#include <math.h>

// ---------------------------------------------------------------------------
// Sinkhorn divergence (GeomLoss-style), N=M=4096, D=64, eps=1, 20 iterations.
// Fused design: never materialize the 4096x4096 cost matrices (201MB > 192MB
// L2).  Each softmin sweep recomputes 2*x_i.y_j tiles on the fp32 WMMA path
// (V_WMMA_F32_16X16X4_F32) and folds them into an online logsumexp.
// Round 2: register-staged B operands (batched loads, staggered waits),
// 4 independent WMMA accumulator chains, 3-exp-per-2-elements online update.
// ---------------------------------------------------------------------------

typedef __attribute__((ext_vector_type(2))) float v2f;
typedef __attribute__((ext_vector_type(8))) float v8f;

#define NPTS 4096
#define DIM 64
#define TILE 16
#define WAVES_PER_BLOCK 8
#define BLOCK_THREADS 256
#define COLS_PER_WAVE (NPTS / WAVES_PER_BLOCK)  // 512
#define TILES_PER_WAVE (COLS_PER_WAVE / TILE)   // 32
#define NEG_BIG (-3.0e38f)

static __device__ __forceinline__ v8f wmma_f32(v2f a, v2f b, v8f c) {
  return __builtin_amdgcn_wmma_f32_16x16x4_f32(
      /*neg_a=*/false, a, /*neg_b=*/false, b,
      /*c_mod=*/(short)0, c, /*reuse_a=*/false, /*reuse_b=*/false);
}

// ---------------------------------------------------------------- reductions
__global__ void sum_ab_kernel(const float* __restrict__ a,
                              const float* __restrict__ b,
                              float* __restrict__ sums) {
  __shared__ float sm0[BLOCK_THREADS];
  __shared__ float sm1[BLOCK_THREADS];
  int t = threadIdx.x;
  float sa = 0.f, sb = 0.f;
  for (int i = t; i < NPTS; i += BLOCK_THREADS) {
    sa += a[i];
    sb += b[i];
  }
  sm0[t] = sa;
  sm1[t] = sb;
  __syncthreads();
  for (int off = BLOCK_THREADS >> 1; off > 0; off >>= 1) {
    if (t < off) {
      sm0[t] += sm0[t + off];
      sm1[t] += sm1[t + off];
    }
    __syncthreads();
  }
  if (t == 0) {
    sums[0] = sm0[0];
    sums[1] = sm1[0];
  }
}

// log-weights, normalized weights, squared norms
__global__ void prep_kernel(const float* __restrict__ x,
                            const float* __restrict__ y,
                            const float* __restrict__ a,
                            const float* __restrict__ b,
                            const float* __restrict__ sums,
                            float* __restrict__ log_a, float* __restrict__ log_b,
                            float* __restrict__ sqnx, float* __restrict__ sqny,
                            float* __restrict__ a_n, float* __restrict__ b_n) {
  int i = blockIdx.x * blockDim.x + threadIdx.x;
  if (i >= NPTS) return;
  float sa = fmaxf(sums[0], 1e-40f);
  float sb = fmaxf(sums[1], 1e-40f);
  float an = a[i] / sa, bn = b[i] / sb;
  a_n[i] = an;
  b_n[i] = bn;
  log_a[i] = __logf(an);
  log_b[i] = __logf(bn);
  const float4* xp = (const float4*)(x + (size_t)i * DIM);
  const float4* yp = (const float4*)(y + (size_t)i * DIM);
  float sx = 0.f, sy = 0.f;
#pragma unroll
  for (int k = 0; k < DIM / 4; ++k) {
    float4 vx = xp[k];
    float4 vy = yp[k];
    sx += vx.x * vx.x + vx.y * vx.y + vx.z * vx.z + vx.w * vx.w;
    sy += vy.x * vy.x + vy.y * vy.y + vy.z * vy.z + vy.w * vy.w;
  }
  sqnx[i] = sx;
  sqny[i] = sy;
}

// ------------------------------------------------------------- softmin sweep
// One launch computes the 4 independent softmins of one Jacobi step.
// blockIdx.y: 0 -> f (C_xy, h=log_b+g)   1 -> g (C_yx, h=log_a+f)
//             2 -> f_aa (C_xx)           3 -> g_bb (C_yy)
// f_i = sqn_row_i - logsumexp_j( h'_j + 2 x_i.y_j ),  h'_j = logw_j+pot_j-sqn_col_j
__global__ void __launch_bounds__(BLOCK_THREADS)
softmin_pass(const float* __restrict__ x, const float* __restrict__ y,
             const float* __restrict__ log_a, const float* __restrict__ log_b,
             const float* __restrict__ sqnx, const float* __restrict__ sqny,
             const float* __restrict__ f_in, const float* __restrict__ g_in,
             const float* __restrict__ faa_in, const float* __restrict__ gbb_in,
             float* __restrict__ f_out, float* __restrict__ g_out,
             float* __restrict__ faa_out, float* __restrict__ gbb_out,
             int avg) {
  __shared__ float hp[NPTS];                            // 16 KB
  __shared__ float red_m[WAVES_PER_BLOCK * 32 * 8];     // 8 KB
  __shared__ float red_s[WAVES_PER_BLOCK * 32 * 8];     // 8 KB

  const int mode = blockIdx.y;
  const float *rows, *cols, *logw, *pot, *colsq, *rowsq, *oldp;
  float* outp;
  switch (mode) {
    case 0:  rows = x; cols = y; logw = log_b; pot = g_in;   colsq = sqny; rowsq = sqnx; oldp = f_in;   outp = f_out;   break;
    case 1:  rows = y; cols = x; logw = log_a; pot = f_in;   colsq = sqnx; rowsq = sqny; oldp = g_in;   outp = g_out;   break;
    case 2:  rows = x; cols = x; logw = log_a; pot = faa_in; colsq = sqnx; rowsq = sqnx; oldp = faa_in; outp = faa_out; break;
    default: rows = y; cols = y; logw = log_b; pot = gbb_in; colsq = sqny; rowsq = sqny; oldp = gbb_in; outp = gbb_out; break;
  }

  // h' vector for all 4096 columns, LDS-resident for the whole sweep
  for (int j = threadIdx.x; j < NPTS; j += BLOCK_THREADS)
    hp[j] = logw[j] + pot[j] - colsq[j];
  __syncthreads();

  const int lane = threadIdx.x & 31;
  const int wave = threadIdx.x >> 5;
  const int rsel = lane & 15;        // M index (A) == N index (B) for this lane
  const int koff = (lane >> 4) * 2;  // lanes 16..31 carry K=2,3 (fp32 16x4 layout)
  const int i0 = blockIdx.x * TILE;
  const float* rowp = rows + (size_t)(i0 + rsel) * DIM + koff;

  // A operand (16x64 row tile): loop-invariant, keep 2x-scaled copy in VGPRs
  v2f a2[16];
#pragma unroll
  for (int kk = 0; kk < 16; ++kk) {
    v2f av = *(const v2f*)(rowp + kk * 4);
    a2[kk] = av + av;  // fold the 2*x.y factor into A
  }

  float m[8], s[8];
#pragma unroll
  for (int r = 0; r < 8; ++r) {
    m[r] = NEG_BIG;
    s[r] = 0.f;
  }

  const int jbase = wave * COLS_PER_WAVE;
  for (int t = 0; t < TILES_PER_WAVE; t += 2) {
    const int j0 = jbase + t * TILE;
    const float* colp0 = cols + (size_t)(j0 + rsel) * DIM + koff;
    const float* colp1 = colp0 + TILE * DIM;
    __builtin_prefetch(colp1 + TILE * DIM, 0, 3);      // tile t+2
    __builtin_prefetch(colp1 + 2 * TILE * DIM, 0, 3);  // tile t+3

    // Stage both B tiles in registers: 32 independent global_load_b64 issued
    // back-to-back -> staggered s_wait_loadcnt instead of load/wait/wmma chains
    v2f b0[16], b1[16];
#pragma unroll
    for (int kk = 0; kk < 16; ++kk) b0[kk] = *(const v2f*)(colp0 + kk * 4);
#pragma unroll
    for (int kk = 0; kk < 16; ++kk) b1[kk] = *(const v2f*)(colp1 + kk * 4);

    // 4 independent accumulator chains (2 tiles x 2 K-halves) for WMMA ILP
    v8f acc00 = {}, acc01 = {}, acc10 = {}, acc11 = {};
#pragma unroll
    for (int kk = 0; kk < 8; ++kk) {
      acc00 = wmma_f32(a2[kk],     b0[kk],     acc00);
      acc10 = wmma_f32(a2[kk],     b1[kk],     acc10);
      acc01 = wmma_f32(a2[kk + 8], b0[kk + 8], acc01);
      acc11 = wmma_f32(a2[kk + 8], b1[kk + 8], acc11);
    }
    v8f acc0 = acc00 + acc01;
    v8f acc1 = acc10 + acc11;

    const float hp0 = hp[j0 + rsel];
    const float hp1 = hp[j0 + TILE + rsel];
#pragma unroll
    for (int r = 0; r < 8; ++r) {  // online logsumexp: 3 exps per 2 elements
      float z0 = acc0[r] + hp0;
      float z1 = acc1[r] + hp1;
      float cm = fmaxf(z0, z1);
      float nm = fmaxf(m[r], cm);
      float e = __expf(z0 - nm) + __expf(z1 - nm);
      s[r] = __builtin_fmaf(s[r], __expf(m[r] - nm), e);
      m[r] = nm;
    }
  }

  const int base = (wave * 32 + lane) * 8;
#pragma unroll
  for (int r = 0; r < 8; ++r) {
    red_m[base + r] = m[r];
    red_s[base + r] = s[r];
  }
  __syncthreads();

  // exact cross-lane / cross-wave logsumexp merge: 128 partials per row
  if (threadIdx.x < TILE) {
    const int rr = threadIdx.x;
    const int r = rr & 7;
    const int half = rr >> 3;  // lanes 0-15 hold M=r, lanes 16-31 hold M=r+8
    float mt = NEG_BIG;
    for (int w = 0; w < WAVES_PER_BLOCK; ++w)
      for (int l = 0; l < 16; ++l)
        mt = fmaxf(mt, red_m[(w * 32 + half * 16 + l) * 8 + r]);
    float st = 0.f;
    for (int w = 0; w < WAVES_PER_BLOCK; ++w)
      for (int l = 0; l < 16; ++l) {
        const int idx = (w * 32 + half * 16 + l) * 8 + r;
        st += red_s[idx] * __expf(red_m[idx] - mt);
      }
    const int i = i0 + rr;
    float val = rowsq[i] - mt - __logf(st);   // f_i = sqn_i - logsumexp
    if (avg) val = 0.5f * (oldp[i] + val);
    outp[i] = val;
  }
}

// ------------------------------------------------------------------- loss
__global__ void loss_kernel(const float* __restrict__ a_n,
                            const float* __restrict__ b_n,
                            const float* __restrict__ f_ab,
                            const float* __restrict__ faa_d,
                            const float* __restrict__ g_ab,
                            const float* __restrict__ gbb_d,
                            float* __restrict__ out) {
  __shared__ float sm[BLOCK_THREADS];
  int t = threadIdx.x;
  float acc = 0.f;
  for (int i = t; i < NPTS; i += BLOCK_THREADS)
    acc += a_n[i] * (f_ab[i] - faa_d[i]) + b_n[i] * (g_ab[i] - gbb_d[i]);
  sm[t] = acc;
  __syncthreads();
  for (int off = BLOCK_THREADS >> 1; off > 0; off >>= 1) {
    if (t < off) sm[t] += sm[t + off];
    __syncthreads();
  }
  if (t == 0) out[0] = sm[0];
}

// ------------------------------------------------------------------- host
extern "C" void kernel_launch(void* const* d_in, const int* in_sizes, int n_in,
                              void* d_out, int out_size, void* d_ws,
                              size_t ws_size, hipStream_t stream) {
  const float* x = (const float*)d_in[0];
  const float* y = (const float*)d_in[1];
  const float* a = (const float*)d_in[2];
  const float* b = (const float*)d_in[3];

  float* ws = (float*)d_ws;
  float* sums = ws;                 // 16 (2 used, padded)
  float* log_a = ws + 16;           // NPTS
  float* log_b = log_a + NPTS;
  float* sqnx = log_b + NPTS;
  float* sqny = sqnx + NPTS;
  float* a_n = sqny + NPTS;
  float* b_n = a_n + NPTS;
  // ping-pong potentials; buffer 0 block is contiguous so one memset zeros it
  float* f0 = b_n + NPTS;
  float* g0 = f0 + NPTS;
  float* faa0 = g0 + NPTS;
  float* gbb0 = faa0 + NPTS;
  float* f1 = gbb0 + NPTS;
  float* g1 = f1 + NPTS;
  float* faa1 = g1 + NPTS;
  float* gbb1 = faa1 + NPTS;
  float* f_ab = gbb1 + NPTS;
  float* g_ab = f_ab + NPTS;
  float* faa_d = g_ab + NPTS;
  float* gbb_d = faa_d + NPTS;

  float* F[2] = {f0, f1};
  float* G[2] = {g0, g1};
  float* FAA[2] = {faa0, faa1};
  float* GBB[2] = {gbb0, gbb1};

  // zero initial potentials (graph-capture-safe)
  hipMemsetAsync(f0, 0, 4 * (size_t)NPTS * sizeof(float), stream);

  sum_ab_kernel<<<1, BLOCK_THREADS, 0, stream>>>(a, b, sums);
  prep_kernel<<<NPTS / BLOCK_THREADS, BLOCK_THREADS, 0, stream>>>(
      x, y, a, b, sums, log_a, log_b, sqnx, sqny, a_n, b_n);

  dim3 grid(NPTS / TILE, 4);
  // init sweep: potentials are zero, no averaging
  softmin_pass<<<grid, BLOCK_THREADS, 0, stream>>>(
      x, y, log_a, log_b, sqnx, sqny, F[0], G[0], FAA[0], GBB[0], F[1], G[1],
      FAA[1], GBB[1], /*avg=*/0);
  int cur = 1;
  // 20 symmetric Jacobi iterations with 0.5 averaging
  for (int it = 0; it < 20; ++it) {
    int nxt = cur ^ 1;
    softmin_pass<<<grid, BLOCK_THREADS, 0, stream>>>(
        x, y, log_a, log_b, sqnx, sqny, F[cur], G[cur], FAA[cur], GBB[cur],
        F[nxt], G[nxt], FAA[nxt], GBB[nxt], /*avg=*/1);
    cur = nxt;
  }
  // final detached softmins
  softmin_pass<<<grid, BLOCK_THREADS, 0, stream>>>(
      x, y, log_a, log_b, sqnx, sqny, F[cur], G[cur], FAA[cur], GBB[cur], f_ab,
      g_ab, faa_d, gbb_d, /*avg=*/0);

  loss_kernel<<<1, BLOCK_THREADS, 0, stream>>>(a_n, b_n, f_ab, faa_d, g_ab,
                                               gbb_d, (float*)d_out);
}